// GaussianMultiViewMerge_49194555408939
// MI455X (gfx1250) — compile-verified
//
#include <hip/hip_runtime.h>
#include <hip/hip_bf16.h>
#include <stdint.h>

// Problem geometry (fixed by the reference).
#define BT      8                      // b*t
#define NELEM   (6 * 96 * 160)         // v*h*w = 92160
#define BN      (BT * NELEM)           // 737280 (divisible by 256)
#define NTRK    512
#define SSEG    (BT * NTRK)            // 4096 segments
#define CH      65                     // summed channels: 3+3+1+3+4+3+32+16
#define THR2    4.0f                   // THR^2
#define DKS     0.05f

typedef __attribute__((ext_vector_type(2))) float v2f;  // native vector for NT stores

// ---- order-preserving float <-> uint encoding for atomic max on floats ----
__device__ __forceinline__ unsigned encf(float f) {
    unsigned b = __float_as_uint(f);
    return (b & 0x80000000u) ? ~b : (b | 0x80000000u);
}
__device__ __forceinline__ float decf(unsigned u) {
    return __uint_as_float((u & 0x80000000u) ? (u & 0x7FFFFFFFu) : ~u);
}

// ---- CDNA5 async global->LDS staging (ASYNCcnt path, CDNA5 ISA) ----
__device__ __forceinline__ void async_copy_b128(const float* g, unsigned lds_byte_off) {
    // VDST = LDS byte offset (wave-relative), VADDR = 64-bit global address, saddr off (GV mode)
    asm volatile("global_load_async_to_lds_b128 %0, %1, off"
                 :: "v"(lds_byte_off), "v"(g)
                 : "memory");
}
__device__ __forceinline__ void wait_async_zero() {
#if __has_builtin(__builtin_amdgcn_s_wait_asynccnt)
    __builtin_amdgcn_s_wait_asynccnt(0);
#else
    asm volatile("s_wait_asynccnt 0" ::: "memory");
#endif
}

// ---------------- Pass 0: init segment scratch ----------------
__global__ __launch_bounds__(256) void k0_init(int* cnt, unsigned* m1k, float* z1,
                                               float* sumec, unsigned* m2k, float* z2,
                                               int* rep, float* ssum) {
    int s = blockIdx.x * 256 + threadIdx.x;
    if (s >= SSEG) return;
    cnt[s] = 0;
    m1k[s] = 0u;          // encf of any finite float is > 0
    m2k[s] = 0u;
    z1[s]  = 0.0f;
    z2[s]  = 0.0f;
    rep[s] = BN;          // sentinel: larger than any element index
    sumec[s * 3 + 0] = 0.0f;
    sumec[s * 3 + 1] = 0.0f;
    sumec[s * 3 + 2] = 0.0f;
    float* p = ssum + (size_t)s * CH;
    #pragma unroll 1
    for (int c = 0; c < CH; ++c) p[c] = 0.0f;
}

// ------- Pass 1 (fused): per-segment counts + m1 = segmax(k0 | valid) -------
// Within a segment all valid elements share cnt, so segmax over valid equals
// segmax over dup on every segment with cnt>=2 (the only ones ever read).
__global__ __launch_bounds__(256) void k1_count_m1(const int* gid, const float* keep,
                                                   int* cnt, unsigned* m1k) {
    int i = blockIdx.x * 256 + threadIdx.x;
    if (i >= BN) return;
    int id = gid[i];
    if (id < 0) return;
    int seg = (i / NELEM) * NTRK + id;
    atomicAdd(&cnt[seg], 1);
    atomicMax(&m1k[seg], encf(keep[i]));
}

// ---------------- Pass 2: z1 and sum(e1 * center) ----------------
__global__ __launch_bounds__(256) void k3_z1(const int* gid, const float* keep,
                                             const float* cen, const int* cnt,
                                             const unsigned* m1k, float* z1, float* sumec) {
    int i = blockIdx.x * 256 + threadIdx.x;
    if (i >= BN) return;
    int id = gid[i];
    if (id < 0) return;
    int seg = (i / NELEM) * NTRK + id;
    if (cnt[seg] < 2) return;
    float e = __expf(keep[i] - decf(m1k[seg]));
    atomicAdd(&z1[seg], e);
    atomicAdd(&sumec[seg * 3 + 0], e * cen[(size_t)i * 3 + 0]);
    atomicAdd(&sumec[seg * 3 + 1], e * cen[(size_t)i * 3 + 1]);
    atomicAdd(&sumec[seg * 3 + 2], e * cen[(size_t)i * 3 + 2]);
}

// ---------------- Pass 3: active mask + m2 = segmax(k0 | active) ----------------
__global__ __launch_bounds__(256) void k4_active(const int* gid, const float* keep,
                                                 const float* cen, const int* cnt,
                                                 const float* z1, const float* sumec,
                                                 unsigned* m2k, unsigned char* act) {
    int i = blockIdx.x * 256 + threadIdx.x;
    if (i >= BN) return;
    unsigned char a = 0;
    int id = gid[i];
    if (id >= 0) {
        int seg = (i / NELEM) * NTRK + id;
        if (cnt[seg] >= 2) {
            float iz = 1.0f / fmaxf(z1[seg], 1e-20f);
            float dx = cen[(size_t)i * 3 + 0] - sumec[seg * 3 + 0] * iz;
            float dy = cen[(size_t)i * 3 + 1] - sumec[seg * 3 + 1] * iz;
            float dz = cen[(size_t)i * 3 + 2] - sumec[seg * 3 + 2] * iz;
            if (dx * dx + dy * dy + dz * dz <= THR2) {
                a = 1;
                atomicMax(&m2k[seg], encf(keep[i]));
            }
        }
    }
    act[i] = a;
}

// ---------------- Pass 4: z2, 65-channel weighted sums, representative ----------------
__global__ __launch_bounds__(256) void k5_accum(const int* gid, const float* keep,
                                                const float* cen, const float* off,
                                                const float* opa, const float* sca,
                                                const float* rot, const float* fdc,
                                                const float* inst, const float* mot,
                                                const unsigned char* act, const unsigned* m2k,
                                                float* z2, float* ssum, int* rep) {
    int i = blockIdx.x * 256 + threadIdx.x;
    if (i >= BN) return;
    if (!act[i]) return;
    int seg = (i / NELEM) * NTRK + gid[i];
    float k0 = keep[i];
    float m2 = decf(m2k[seg]);
    float e  = __expf(k0 - m2);
    atomicAdd(&z2[seg], e);
    float* s = ssum + (size_t)seg * CH;
    #pragma unroll
    for (int c = 0; c < 3; ++c) atomicAdd(&s[c],      e * cen[(size_t)i * 3 + c]);
    #pragma unroll
    for (int c = 0; c < 3; ++c) atomicAdd(&s[3 + c],  e * off[(size_t)i * 3 + c]);
    atomicAdd(&s[6], e * opa[i]);
    #pragma unroll
    for (int c = 0; c < 3; ++c) atomicAdd(&s[7 + c],  e * sca[(size_t)i * 3 + c]);
    #pragma unroll
    for (int c = 0; c < 4; ++c) atomicAdd(&s[10 + c], e * rot[(size_t)i * 4 + c]);
    #pragma unroll
    for (int c = 0; c < 3; ++c) atomicAdd(&s[14 + c], e * fdc[(size_t)i * 3 + c]);
    #pragma unroll 4
    for (int c = 0; c < 32; ++c) atomicAdd(&s[17 + c], e * inst[(size_t)i * 32 + c]);
    #pragma unroll 4
    for (int c = 0; c < 16; ++c) atomicAdd(&s[49 + c], e * mot[(size_t)i * 16 + c]);
    if (k0 == m2) atomicMin(&rep[seg], i);
}

// ---------------- Pass 5: finalize segments (scale by 1/z2, normalize quaternion) ----
__global__ __launch_bounds__(256) void k6_final(float* z2, float* ssum) {
    int s = blockIdx.x * 256 + threadIdx.x;
    if (s >= SSEG) return;
    float iz = 1.0f / fmaxf(z2[s], 1e-20f);
    float* p = ssum + (size_t)s * CH;
    #pragma unroll 1
    for (int c = 0; c < CH; ++c) p[c] *= iz;
    float r0 = p[10], r1 = p[11], r2 = p[12], r3 = p[13];
    float n = fmaxf(sqrtf(r0 * r0 + r1 * r1 + r2 * r2 + r3 * r3), 1e-12f);
    p[10] = r0 / n; p[11] = r1 / n; p[12] = r2 / n; p[13] = r3 / n;
}

// ---------------- Pass 6: output select ----------------
// Passthrough lanes stage inst(128B)+motion(64B) via the CDNA5 async data mover
// (ASYNCcnt path); merged lanes read only the tiny L2-resident segment table.
// Output is a write-once 195MB stream (> 192MB L2) -> non-temporal stores.
__global__ __launch_bounds__(256) void k7_output(const int* gid, const float* keep,
                                                 const float* cen, const float* off,
                                                 const float* opa, const float* sca,
                                                 const float* rot, const float* fdc,
                                                 const float* inst, const float* mot,
                                                 const unsigned char* act, const float* ssum,
                                                 const unsigned* m2k, const int* rep,
                                                 float* out) {
    __shared__ float sh[256 * 48];            // 48 KB: 32 inst + 16 motion floats per thread
    int t = threadIdx.x;
    int i = blockIdx.x * 256 + t;             // BN % 256 == 0
    float o[66];
    bool a = act[i] != 0;
    if (a) {
        int seg = (i / NELEM) * NTRK + gid[i];
        const float* s = ssum + (size_t)seg * CH;
        float sf = (i == rep[seg]) ? 1.0f : DKS;
        #pragma unroll
        for (int c = 0; c < 3; ++c) o[c]      = s[c];        // mcen
        #pragma unroll
        for (int c = 0; c < 3; ++c) o[3 + c]  = s[3 + c];    // moff
        o[6] = s[6] * sf;                                    // mop * sfac
        #pragma unroll
        for (int c = 0; c < 3; ++c) o[7 + c]  = s[7 + c];    // msc
        #pragma unroll
        for (int c = 0; c < 4; ++c) o[10 + c] = s[10 + c];   // mrot (normalized)
        #pragma unroll
        for (int c = 0; c < 3; ++c) o[14 + c] = s[14 + c];   // mfd
        o[17] = decf(m2k[seg]) * sf;                         // mkeep * sfac
        #pragma unroll 8
        for (int c = 0; c < 32; ++c) o[18 + c] = s[17 + c];  // minst
        #pragma unroll 8
        for (int c = 0; c < 16; ++c) o[50 + c] = s[49 + c];  // mmot
    } else {
        // Kick off async staging first, overlap with the 18 scalar passthrough loads.
        unsigned lbase = (unsigned)(uintptr_t)(const void*)&sh[t * 48];
        const float* gi = inst + (size_t)i * 32;  // 128 B, 16B-aligned chunks
        const float* gm = mot  + (size_t)i * 16;  // 64 B
        #pragma unroll
        for (int q = 0; q < 8; ++q) async_copy_b128(gi + q * 4, lbase + (unsigned)(q * 16));
        #pragma unroll
        for (int q = 0; q < 4; ++q) async_copy_b128(gm + q * 4, lbase + (unsigned)(128 + q * 16));

        #pragma unroll
        for (int c = 0; c < 3; ++c) o[c]      = cen[(size_t)i * 3 + c];
        #pragma unroll
        for (int c = 0; c < 3; ++c) o[3 + c]  = off[(size_t)i * 3 + c];
        o[6] = opa[i];
        #pragma unroll
        for (int c = 0; c < 3; ++c) o[7 + c]  = sca[(size_t)i * 3 + c];
        #pragma unroll
        for (int c = 0; c < 4; ++c) o[10 + c] = rot[(size_t)i * 4 + c];
        #pragma unroll
        for (int c = 0; c < 3; ++c) o[14 + c] = fdc[(size_t)i * 3 + c];
        o[17] = keep[i];
        wait_async_zero();                        // ASYNCcnt==0 -> staged LDS data visible
        #pragma unroll 8
        for (int c = 0; c < 32; ++c) o[18 + c] = sh[t * 48 + c];
        #pragma unroll 8
        for (int c = 0; c < 16; ++c) o[50 + c] = sh[t * 48 + 32 + c];
    }
    // 264 B/row, 8B-aligned -> stream out as 33 x b64 non-temporal stores.
    v2f* po2 = reinterpret_cast<v2f*>(out + (size_t)i * 66);
    #pragma unroll 11
    for (int c = 0; c < 33; ++c) {
        v2f v;
        v.x = o[2 * c];
        v.y = o[2 * c + 1];
        __builtin_nontemporal_store(v, po2 + c);
    }
}

extern "C" void kernel_launch(void* const* d_in, const int* in_sizes, int n_in,
                              void* d_out, int out_size, void* d_ws, size_t ws_size,
                              hipStream_t stream) {
    (void)in_sizes; (void)n_in; (void)out_size; (void)ws_size;
    const float* cen  = (const float*)d_in[0];
    const float* off  = (const float*)d_in[1];
    const float* opa  = (const float*)d_in[2];
    const float* sca  = (const float*)d_in[3];
    const float* rot  = (const float*)d_in[4];
    const float* fdc  = (const float*)d_in[5];
    const float* keep = (const float*)d_in[6];
    const float* inst = (const float*)d_in[7];
    const float* mot  = (const float*)d_in[8];
    const int*   gid  = (const int*)d_in[9];
    float* out = (float*)d_out;

    // Workspace layout (all re-initialized every call; ~1.95 MB, L2-resident).
    int*      cnt   = (int*)d_ws;                       // SSEG
    unsigned* m1k   = (unsigned*)(cnt + SSEG);          // SSEG
    float*    z1    = (float*)(m1k + SSEG);             // SSEG
    float*    sumec = z1 + SSEG;                        // 3*SSEG
    unsigned* m2k   = (unsigned*)(sumec + 3 * SSEG);    // SSEG
    float*    z2    = (float*)(m2k + SSEG);             // SSEG
    int*      rep   = (int*)(z2 + SSEG);                // SSEG
    float*    ssum  = (float*)(rep + SSEG);             // CH*SSEG
    unsigned char* act = (unsigned char*)(ssum + (size_t)CH * SSEG); // BN bytes

    dim3 blk(256);
    dim3 gSeg((SSEG + 255) / 256);
    dim3 gEl(BN / 256);

    k0_init    <<<gSeg, blk, 0, stream>>>(cnt, m1k, z1, sumec, m2k, z2, rep, ssum);
    k1_count_m1<<<gEl,  blk, 0, stream>>>(gid, keep, cnt, m1k);
    k3_z1      <<<gEl,  blk, 0, stream>>>(gid, keep, cen, cnt, m1k, z1, sumec);
    k4_active  <<<gEl,  blk, 0, stream>>>(gid, keep, cen, cnt, z1, sumec, m2k, act);
    k5_accum   <<<gEl,  blk, 0, stream>>>(gid, keep, cen, off, opa, sca, rot, fdc,
                                          inst, mot, act, m2k, z2, ssum, rep);
    k6_final   <<<gSeg, blk, 0, stream>>>(z2, ssum);
    k7_output  <<<gEl,  blk, 0, stream>>>(gid, keep, cen, off, opa, sca, rot, fdc,
                                          inst, mot, act, ssum, m2k, rep, out);
}